// ParametricEqualizer_21586505630173
// MI455X (gfx1250) — compile-verified
//
#include <hip/hip_runtime.h>

// Peaking equalizer: [B=64, P=256] params -> [16384, 1025] complex64 response.
// H(k) = (b0 + b1 e^{-i t} + b2 e^{-2i t}) / (a0 + a1 e^{-i t} + a2 e^{-2i t}),
// t = 2*pi*k/2048. The 3-tap DFT is done as WMMA F32 16x16x4 (K padded to 4).

typedef __attribute__((ext_vector_type(2))) float v2f;
typedef __attribute__((ext_vector_type(8))) float v8f;

#define PEQ_SR      22050.0f
#define PEQ_NFFT    2048
#define PEQ_NBINS   1025          // NFFT/2 + 1
#define PEQ_NFILT   16384         // 64 * 256
#define PEQ_TILES_F (PEQ_NFILT / 16)          // 1024
#define PEQ_TILES_K ((PEQ_NBINS + 15) / 16)   // 65
#define PEQ_WAVES_PER_BLOCK 8
#define PEQ_NBLOCKS (PEQ_TILES_F * PEQ_TILES_K / PEQ_WAVES_PER_BLOCK) // 8320

__global__ __launch_bounds__(256)
void peq_wmma_kernel(const float* __restrict__ center,
                     const float* __restrict__ gain,
                     const float* __restrict__ q,
                     float* __restrict__ out)
{
    const int lane  = threadIdx.x & 31;
    const int wave  = threadIdx.x >> 5;
    const int wtask = blockIdx.x * PEQ_WAVES_PER_BLOCK + wave;   // wave-uniform
    const int tile_f = wtask / PEQ_TILES_K;                      // filter tile
    const int tile_k = wtask % PEQ_TILES_K;                      // bin tile

    const int  sub = lane & 15;       // row (A) / col (B) within 16
    const bool hi  = lane >= 16;      // second K-pair half of the wave

    // ---- A matrices (16 filters x K=4 taps), per ISA 32-bit A 16x4 layout:
    // VGPR0: lanes 0-15 K=0, lanes 16-31 K=2 ; VGPR1: lanes 0-15 K=1, lanes 16-31 K=3
    const int fbase = tile_f * 16;
    const int fidx  = fbase + sub;
    const float c  = center[fidx];
    const float g  = gain[fidx];
    const float qq = q[fidx];

    const float w0 = 6.28318530717958647692f * c / PEQ_SR;
    float sw0, cw0;
    __sincosf(w0, &sw0, &cw0);
    const float alpha = sw0 / (2.0f * qq);
    // A_amp = 10^(gain/40) = exp(gain * ln(10)/40)
    const float Aamp  = __expf(g * 0.05756462732485114210f);
    const float t1    = -2.0f * cw0;                 // shared middle tap
    const float nb0   = 1.0f + alpha * Aamp;         // numerator taps (fir)
    const float nb2   = 1.0f - alpha * Aamp;
    const float da0   = 1.0f + alpha / Aamp;         // denominator taps (iir)
    const float da2   = 1.0f - alpha / Aamp;

    v2f Anum, Aden;
    Anum.x = hi ? nb2 : nb0;   Anum.y = hi ? 0.0f : t1;
    Aden.x = hi ? da2 : da0;   Aden.y = hi ? 0.0f : t1;

    // ---- B matrices (K=4 x 16 bins): twiddles. Mirror layout:
    // VGPR0: lanes 0-15 K=0, lanes 16-31 K=2 ; VGPR1: lanes 0-15 K=1, lanes 16-31 K=3
    const int k = tile_k * 16 + sub;                 // frequency bin (col)
    const float theta = 6.28318530717958647692f * (float)k / (float)PEQ_NFFT;
    float s1, c1;
    __sincosf(theta, &s1, &c1);
    const float c2 = 2.0f * c1 * c1 - 1.0f;          // cos 2θ
    const float s2 = 2.0f * s1 * c1;                 // sin 2θ

    v2f Bcos, Bsin;
    Bcos.x = hi ? c2    : 1.0f;   Bcos.y = hi ? 0.0f :  c1;
    Bsin.x = hi ? -s2   : 0.0f;   Bsin.y = hi ? 0.0f : -s1;

    // ---- 4 WMMAs: Re/Im of numerator & denominator, identical C/D layouts
    v8f zero = {};
    v8f nr = __builtin_amdgcn_wmma_f32_16x16x4_f32(false, Anum, false, Bcos,
                                                   (short)0, zero, false, false);
    v8f ni = __builtin_amdgcn_wmma_f32_16x16x4_f32(false, Anum, false, Bsin,
                                                   (short)0, zero, false, false);
    v8f dr = __builtin_amdgcn_wmma_f32_16x16x4_f32(false, Aden, false, Bcos,
                                                   (short)0, zero, false, false);
    v8f di = __builtin_amdgcn_wmma_f32_16x16x4_f32(false, Aden, false, Bsin,
                                                   (short)0, zero, false, false);

    // ---- lane-local complex division + streaming (non-temporal) stores.
    // C/D layout: lane L, VGPR v -> (M = v + 8*(L>=16), N = L%16)
    if (k < PEQ_NBINS) {                 // only Nyquist tile partially stores
        const int mofs = hi ? 8 : 0;
        v2f* o2 = (v2f*)out;
#pragma unroll
        for (int v = 0; v < 8; ++v) {
            const int f = fbase + mofs + v;
            const float NR = nr[v], NI = ni[v], DR = dr[v], DI = di[v];
            const float inv = __builtin_amdgcn_rcpf(DR * DR + DI * DI);
            v2f res;
            res.x = (NR * DR + NI * DI) * inv;   // real
            res.y = (NI * DR - NR * DI) * inv;   // imag
            const size_t off = (size_t)f * PEQ_NBINS + (size_t)k;
            __builtin_nontemporal_store(res, &o2[off]);
        }
    }
}

extern "C" void kernel_launch(void* const* d_in, const int* in_sizes, int n_in,
                              void* d_out, int out_size, void* d_ws, size_t ws_size,
                              hipStream_t stream) {
    const float* center = (const float*)d_in[0];
    const float* gain   = (const float*)d_in[1];
    const float* q      = (const float*)d_in[2];
    float* out = (float*)d_out;

    dim3 grid(PEQ_NBLOCKS), block(256);
    hipLaunchKernelGGL(peq_wmma_kernel, grid, block, 0, stream,
                       center, gain, q, out);
}